// GATNet_28484223107177
// MI455X (gfx1250) — compile-verified
//
#include <hip/hip_runtime.h>
#include <hip/hip_bf16.h>

typedef __attribute__((ext_vector_type(2))) float v2f;
typedef __attribute__((ext_vector_type(8))) float v8f;

#define NEG_SLOPE 0.2f

// ---------------------------------------------------------------------------
// Order-preserving float <-> uint encoding so segment-max can use the
// single hardware GLOBAL_ATOMIC_MAX_U32 (no CAS loop).
// ---------------------------------------------------------------------------
__device__ __forceinline__ unsigned enc_f(float f) {
    unsigned u = __float_as_uint(f);
    return (u & 0x80000000u) ? ~u : (u | 0x80000000u);
}
__device__ __forceinline__ float dec_f(unsigned e) {
    unsigned u = (e & 0x80000000u) ? (e & 0x7FFFFFFFu) : ~e;
    return __uint_as_float(u);
}

// ---------------------------------------------------------------------------
// Node GEMM + fused attention logits, one 16-node tile per wave32 using
// V_WMMA_F32_16X16X4_F32.  H[n,:] = X[n,:] @ W ;  aS[n] = H[n,:]·avS ; etc.
//   A (16x4 f32): lanes 0-15 hold K={0,1}, lanes 16-31 hold K={2,3}, M=lane%16
//   B (4x16 f32): row K striped across lanes (N=lane%16), K = 2*(lane/16)+vgpr
//   C/D (16x16):  lane holds column N=lane%16, VGPR r -> row M = r + 8*(lane/16)
// All tile loads are branch-free: indices are clamped to valid addresses and
// out-of-range contributions zeroed with selects (v_cndmask), so no EXEC
// save/restore sequences fence the operand loads between WMMAs.
// ---------------------------------------------------------------------------
template <int KDIM, int FDIM>
__global__ void node_gemm_alpha(const float* __restrict__ X,
                                const float* __restrict__ W,     // KDIM x FDIM row-major
                                const float* __restrict__ avS,   // [FDIM]
                                const float* __restrict__ avD,   // [FDIM]
                                float* __restrict__ H,           // [n, FDIM]
                                float* __restrict__ aS,          // [n]
                                float* __restrict__ aD,          // [n]
                                int nNodes) {
    constexpr int NT = FDIM / 16;           // 1 or 2 column tiles
    const int lane  = threadIdx.x & 31;
    const int wave  = threadIdx.x >> 5;
    const int tile  = blockIdx.x * (blockDim.x >> 5) + wave;
    if (tile * 16 >= nNodes) return;        // wave-uniform exit (EXEC full at WMMA)

    const int  mrow  = lane & 15;           // A row this lane feeds
    const int  khalf = lane >> 4;           // 0: K={0,1}, 1: K={2,3}
    const int  col   = lane & 15;           // B/C/D column within tile
    const int  node  = tile * 16 + mrow;
    const bool mv    = node < nNodes;
    const int  nodeC = mv ? node : (nNodes - 1);   // clamped, always valid
    const float* __restrict__ Xrow = X + (size_t)nodeC * KDIM;

    v8f c[NT] = {};

#pragma unroll
    for (int kb = 0; kb < KDIM; kb += 4) {
        const int  k0  = kb + 2 * khalf;
        const int  k1  = k0 + 1;
        const bool v0  = (k0 < KDIM);
        const bool v1  = (k1 < KDIM);
        const int  k0c = v0 ? k0 : 0;
        const int  k1c = v1 ? k1 : 0;
        // Unconditional loads from clamped addresses; zero via select.
        const float ax = Xrow[k0c];
        const float ay = Xrow[k1c];
        v2f a;
        a.x = (v0 && mv) ? ax : 0.0f;
        a.y = (v1 && mv) ? ay : 0.0f;
#pragma unroll
        for (int t = 0; t < NT; t++) {
            const int n = t * 16 + col;
            const float bx = W[k0c * FDIM + n];
            const float by = W[k1c * FDIM + n];
            v2f b;
            b.x = v0 ? bx : 0.0f;
            b.y = v1 ? by : 0.0f;
            c[t] = __builtin_amdgcn_wmma_f32_16x16x4_f32(
                false, a, false, b, (short)0, c[t], false, false);
        }
    }

    // Store H tile and reduce attention logits across the 16 lanes of a row.
#pragma unroll
    for (int r = 0; r < 8; r++) {
        const int  nodeR = tile * 16 + r + 8 * khalf;
        const bool rv    = nodeR < nNodes;
        float ps = 0.0f, pd = 0.0f;
#pragma unroll
        for (int t = 0; t < NT; t++) {
            const float v = c[t][r];
            const int n = t * 16 + col;
            if (rv) H[(size_t)nodeR * FDIM + n] = v;
            ps += v * avS[n];
            pd += v * avD[n];
        }
#pragma unroll
        for (int off = 8; off >= 1; off >>= 1) {   // stays within 16-lane half
            ps += __shfl_xor(ps, off, 32);
            pd += __shfl_xor(pd, off, 32);
        }
        if (col == 0 && rv) { aS[nodeR] = ps; aD[nodeR] = pd; }
    }
}

// ---------------------------------------------------------------------------
// Edge pass 1: e = leaky_relu(aS[src] + aD[dst]); stash e; segment-max(dst).
// Edge i >= E is the self-loop for node i-E.
// ---------------------------------------------------------------------------
__global__ void edge_pass1(const int* __restrict__ ei, long long E_, int N_,
                           const float* __restrict__ aS, const float* __restrict__ aD,
                           float* __restrict__ ews, unsigned* __restrict__ menc) {
    const long long i = (long long)blockIdx.x * blockDim.x + threadIdx.x;
    const long long ET = E_ + N_;
    if (i >= ET) return;
    int s, d;
    if (i < E_) { s = ei[i]; d = ei[E_ + i]; }
    else        { s = d = (int)(i - E_); }
    float e = aS[s] + aD[d];
    e = e > 0.0f ? e : NEG_SLOPE * e;
    ews[i] = e;
    atomicMax(&menc[d], enc_f(e));
}

// ---------------------------------------------------------------------------
// Edge pass 2 (fused): ex = exp(e - m[dst]); s[dst] += ex;
//                      acc[dst,:] += ex * H[src,:]   (unnormalized).
// gridDim.y = FDIM/8 feature chunks; chunk 0 also accumulates the softmax sum.
// All gather/scatter targets live in the 192MB L2.
// ---------------------------------------------------------------------------
template <int FDIM>
__global__ void edge_pass2(const int* __restrict__ ei, long long E_, int N_,
                           const float* __restrict__ ews,
                           const unsigned* __restrict__ menc,
                           const float* __restrict__ H,
                           float* __restrict__ ssum, float* __restrict__ acc) {
    const long long i = (long long)blockIdx.x * blockDim.x + threadIdx.x;
    const long long ET = E_ + N_;
    if (i >= ET) return;
    const int c = blockIdx.y;                 // feature chunk of 8
    int s, d;
    if (i < E_) { s = ei[i]; d = ei[E_ + i]; }
    else        { s = d = (int)(i - E_); }
    const float m  = dec_f(menc[d]);
    const float ex = __expf(ews[i] - m);
    if (c == 0) atomicAdd(&ssum[d], ex);
    const float* hs = H   + (size_t)s * FDIM + c * 8;
    float*       ac = acc + (size_t)d * FDIM + c * 8;
    const float4 h0 = *(const float4*)(hs);
    const float4 h1 = *(const float4*)(hs + 4);
    atomicAdd(ac + 0, ex * h0.x);
    atomicAdd(ac + 1, ex * h0.y);
    atomicAdd(ac + 2, ex * h0.z);
    atomicAdd(ac + 3, ex * h0.w);
    atomicAdd(ac + 4, ex * h1.x);
    atomicAdd(ac + 5, ex * h1.y);
    atomicAdd(ac + 6, ex * h1.z);
    atomicAdd(ac + 7, ex * h1.w);
}

// ---------------------------------------------------------------------------
// Per-node finish of a GAT layer: g = relu(acc / s + b), in place.
// ---------------------------------------------------------------------------
__global__ void node_finish(float* __restrict__ acc, const float* __restrict__ ssum,
                            const float* __restrict__ bias, int N_, int FDIM) {
    const long long i = (long long)blockIdx.x * blockDim.x + threadIdx.x;
    if (i >= (long long)N_ * FDIM) return;
    const int n = (int)(i / FDIM);
    const int f = (int)(i % FDIM);
    const float v = acc[i] / ssum[n] + bias[f];
    acc[i] = fmaxf(v, 0.0f);
}

// ---------------------------------------------------------------------------
// Final: out[n,:] = relu(acc2[n,:]/s[n] + b2) @ Wout + bout   (32 -> 4)
// ---------------------------------------------------------------------------
__global__ void final_out(const float* __restrict__ acc2, const float* __restrict__ ssum,
                          const float* __restrict__ b2, const float* __restrict__ Wout,
                          const float* __restrict__ bout, float* __restrict__ out, int N_) {
    const int n = blockIdx.x * blockDim.x + threadIdx.x;
    if (n >= N_) return;
    float o0 = bout[0], o1 = bout[1], o2 = bout[2], o3 = bout[3];
    const float inv = 1.0f / ssum[n];
#pragma unroll
    for (int f = 0; f < 32; f++) {
        const float g = fmaxf(acc2[(size_t)n * 32 + f] * inv + b2[f], 0.0f);
        o0 += g * Wout[f * 4 + 0];
        o1 += g * Wout[f * 4 + 1];
        o2 += g * Wout[f * 4 + 2];
        o3 += g * Wout[f * 4 + 3];
    }
    *(float4*)(out + (size_t)n * 4) = make_float4(o0, o1, o2, o3);
}

// ---------------------------------------------------------------------------
extern "C" void kernel_launch(void* const* d_in, const int* in_sizes, int n_in,
                              void* d_out, int out_size, void* d_ws, size_t ws_size,
                              hipStream_t stream) {
    const int       N_ = in_sizes[0] / 7;       // x: [N,7]
    const long long E_ = in_sizes[1] / 2;       // edge_index: [2,E]
    const long long ET = E_ + N_;

    const float* x    = (const float*)d_in[0];
    const int*   ei   = (const int*)d_in[1];
    // d_in[2] = edge_attr (ignored: GATConv with edge_dim=None)
    const float* W1   = (const float*)d_in[3];
    const float* as1v = (const float*)d_in[4];
    const float* ad1v = (const float*)d_in[5];
    const float* b1   = (const float*)d_in[6];
    const float* W2   = (const float*)d_in[7];
    const float* as2v = (const float*)d_in[8];
    const float* ad2v = (const float*)d_in[9];
    const float* b2   = (const float*)d_in[10];
    const float* Wo   = (const float*)d_in[11];
    const float* bo   = (const float*)d_in[12];

    float* ws = (float*)d_ws;
    size_t o = 0;
    float*    h1   = ws + o;              o += (size_t)N_ * 16;
    float*    acc1 = ws + o;              o += (size_t)N_ * 16;  // becomes g1
    float*    h2   = ws + o;              o += (size_t)N_ * 32;
    float*    acc2 = ws + o;              o += (size_t)N_ * 32;
    float*    aS   = ws + o;              o += (size_t)N_;
    float*    aD   = ws + o;              o += (size_t)N_;
    unsigned* menc = (unsigned*)(ws + o); o += (size_t)N_;
    float*    ssum = ws + o;              o += (size_t)N_;
    float*    ews  = ws + o;              o += (size_t)ET;

    const int nTiles = (N_ + 15) / 16;
    const dim3 gGemm((nTiles + 7) / 8);          // 8 waves (tiles) per 256-thread block
    const dim3 gE1((unsigned)((ET + 255) / 256));
    const dim3 gE2a((unsigned)((ET + 255) / 256), 2);   // F=16 -> 2 chunks
    const dim3 gE2b((unsigned)((ET + 255) / 256), 4);   // F=32 -> 4 chunks
    const dim3 gNF1((unsigned)(((long long)N_ * 16 + 255) / 256));
    const dim3 gOut((unsigned)((N_ + 255) / 256));

    // ---- Layer 1 ----
    hipMemsetAsync(menc, 0, (size_t)N_ * 4, stream);                 // enc(-inf) floor
    hipMemsetAsync(ssum, 0, (size_t)N_ * 4, stream);
    hipMemsetAsync(acc1, 0, (size_t)N_ * 16 * 4, stream);
    node_gemm_alpha<7, 16><<<gGemm, 256, 0, stream>>>(x, W1, as1v, ad1v,
                                                      h1, aS, aD, N_);
    edge_pass1<<<gE1, 256, 0, stream>>>(ei, E_, N_, aS, aD, ews, menc);
    edge_pass2<16><<<gE2a, 256, 0, stream>>>(ei, E_, N_, ews, menc, h1, ssum, acc1);
    node_finish<<<gNF1, 256, 0, stream>>>(acc1, ssum, b1, N_, 16);   // acc1 -> g1

    // ---- Layer 2 ----
    hipMemsetAsync(menc, 0, (size_t)N_ * 4, stream);
    hipMemsetAsync(ssum, 0, (size_t)N_ * 4, stream);
    hipMemsetAsync(acc2, 0, (size_t)N_ * 32 * 4, stream);
    node_gemm_alpha<16, 32><<<gGemm, 256, 0, stream>>>(acc1, W2, as2v, ad2v,
                                                       h2, aS, aD, N_);
    edge_pass1<<<gE1, 256, 0, stream>>>(ei, E_, N_, aS, aD, ews, menc);
    edge_pass2<32><<<gE2b, 256, 0, stream>>>(ei, E_, N_, ews, menc, h2, ssum, acc2);

    // ---- Output projection ----
    final_out<<<gOut, 256, 0, stream>>>(acc2, ssum, b2, Wo, bo, (float*)d_out, N_);
}